// SparseBMM_23940147708357
// MI455X (gfx1250) — compile-verified
//
#include <hip/hip_runtime.h>
#include <hip/hip_bf16.h>

typedef __attribute__((ext_vector_type(2))) float v2f;
typedef __attribute__((ext_vector_type(4))) float v4f;
typedef __attribute__((ext_vector_type(8))) float v8f;

#define MK      2048
#define NBATCH  8
#define TILE64  64
#define NT      (MK / TILE64)   // 32 activity tiles per dim
#define THRESH  1e-6f

// GEMM tiling
#define BM 256
#define BN 128
#define BK 32
#define LDA (BK + 2)   // 34 floats: even (8B-aligned b64) + bank-conflict-free
#define LDB (BK + 2)

// ---------------------------------------------------------------------------
// Pass 1: activity prescan. One 256-thread block per (b, mt, kt) 64x64 A-tile.
// flags[b][mt][kt] = (max|a| > THRESH) ? 1.0f : 0.0f
// ---------------------------------------------------------------------------
__global__ __launch_bounds__(256) void prescan_activity_kernel(
    const float* __restrict__ a, float* __restrict__ flags) {
  const int kt = blockIdx.x, mt = blockIdx.y, b = blockIdx.z;
  const int tid = threadIdx.x;
  const float* base =
      a + (size_t)b * MK * MK + (size_t)(mt * TILE64) * MK + (size_t)kt * TILE64;

  float m = 0.0f;
#pragma unroll
  for (int i = 0; i < 16; ++i) {               // 4096 elems / 256 threads
    int e = tid + i * 256;
    int r = e >> 6, c = e & 63;
    m = fmaxf(m, __builtin_fabsf(base[(size_t)r * MK + c]));
  }
  // wave32 butterfly reduce
#pragma unroll
  for (int ofs = 16; ofs > 0; ofs >>= 1)
    m = fmaxf(m, __shfl_xor(m, ofs, 32));

  __shared__ float red[8];
  if ((tid & 31) == 0) red[tid >> 5] = m;
  __syncthreads();
  if (tid == 0) {
    float mm = red[0];
#pragma unroll
    for (int i = 1; i < 8; ++i) mm = fmaxf(mm, red[i]);
    flags[((size_t)b * NT + mt) * NT + kt] = (mm > THRESH) ? 1.0f : 0.0f;
  }
}

// ---------------------------------------------------------------------------
// Pass 2: tiled f32 WMMA GEMM. Workgroup (8 waves) computes a 256x128 C tile;
// each wave computes 64x64 (4x4 subtiles of 16x16) via V_WMMA_F32_16X16X4_F32.
// ---------------------------------------------------------------------------
__global__ __launch_bounds__(256) void sparse_bmm_wmma_kernel(
    const float* __restrict__ A, const float* __restrict__ Bm,
    const float* __restrict__ flags, float* __restrict__ C) {
  __shared__ float As[BM * LDA];   // [row][k], row-major, stride 34
  __shared__ float Bs[BN * LDB];   // [n][k]  (B transposed), stride 34

  const int b   = blockIdx.z;
  const int m0  = blockIdx.y * BM;
  const int n0  = blockIdx.x * BN;
  const int tid = threadIdx.x;
  const int lane = tid & 31;
  const int wave = tid >> 5;
  const int wm = (wave & 3) * 64;   // wave's M offset inside C tile (4 waves)
  const int wn = (wave >> 2) * 64;  // wave's N offset inside C tile (2 waves)

  const size_t boff = (size_t)b * MK * MK;
  const float* __restrict__ Ag = A  + boff;
  const float* __restrict__ Bg = Bm + boff;
  float*       __restrict__ Cg = C  + boff;
  const float* __restrict__ Fg = flags + (size_t)b * NT * NT;

  const v8f vzero = {};
  v8f acc[4][4];
#pragma unroll
  for (int i = 0; i < 4; ++i)
#pragma unroll
    for (int j = 0; j < 4; ++j) acc[i][j] = vzero;

  // global->LDS mapping
  const int arow = tid >> 3;         // 0..31 (x8 iters -> 256 rows)
  const int acol = (tid & 7) << 2;   // 0..28, float4 along k (contiguous 128B/8 lanes)
  const int brow = tid >> 5;         // 0..7  (x4 iters -> 32 k-rows)
  const int bcol = lane << 2;        // 0..124, float4 along n

  // WMMA f32 fragment addressing: lane L holds row (L&15), k pair base +2*(L>>4)
  const int rsel = lane & 15;
  const int ksel = (lane >> 4) << 1;

  for (int k0 = 0; k0 < MK; k0 += BK) {
    // ---- A: 256x32, masked by activity flag ----
#pragma unroll
    for (int it = 0; it < 8; ++it) {
      int row = arow + it * 32;
      float flag = Fg[(size_t)((m0 + row) >> 6) * NT + ((k0 + acol) >> 6)];
      v4f v = *(const v4f*)(Ag + (size_t)(m0 + row) * MK + (k0 + acol));
      v *= flag;
      *(v4f*)(&As[row * LDA + acol]) = v;
    }
    // ---- B: 32x128, stored transposed [n][k] ----
#pragma unroll
    for (int it = 0; it < 4; ++it) {
      int kr = brow + it * 8;
      v4f v = *(const v4f*)(Bg + (size_t)(k0 + kr) * MK + (n0 + bcol));
      Bs[(bcol + 0) * LDB + kr] = v.x;
      Bs[(bcol + 1) * LDB + kr] = v.y;
      Bs[(bcol + 2) * LDB + kr] = v.z;
      Bs[(bcol + 3) * LDB + kr] = v.w;
    }
    // prefetch next K block (speculative; dropped if OOB-translated)
    if (k0 + BK < MK) {
      __builtin_prefetch(Ag + (size_t)(m0 + arow) * MK + (k0 + BK + acol), 0, 1);
      __builtin_prefetch(Bg + (size_t)(k0 + BK + brow) * MK + (n0 + bcol), 0, 1);
    }
    __syncthreads();

    // ---- compute: 8 k-steps x 16 WMMA, 8 ds_load_b64 per k-step ----
#pragma unroll
    for (int kk = 0; kk < BK; kk += 4) {
      v2f af[4], bf[4];
#pragma unroll
      for (int i = 0; i < 4; ++i)
        af[i] = *(const v2f*)(&As[(wm + i * 16 + rsel) * LDA + kk + ksel]);
#pragma unroll
      for (int j = 0; j < 4; ++j)
        bf[j] = *(const v2f*)(&Bs[(wn + j * 16 + rsel) * LDB + kk + ksel]);
#pragma unroll
      for (int i = 0; i < 4; ++i)
#pragma unroll
        for (int j = 0; j < 4; ++j)
          acc[i][j] = __builtin_amdgcn_wmma_f32_16x16x4_f32(
              false, af[i], false, bf[j], (short)0, acc[i][j], false, false);
    }
    __syncthreads();
  }

  // ---- store: C/D layout: VGPR r -> M=r (lanes 0-15) / M=r+8 (lanes 16-31),
  //      N = lane&15 ----
  const int nl = lane & 15;
  const int mh = (lane >> 4) << 3;  // 0 or 8
#pragma unroll
  for (int i = 0; i < 4; ++i)
#pragma unroll
    for (int j = 0; j < 4; ++j) {
      const size_t cbase =
          (size_t)(m0 + wm + i * 16 + mh) * MK + (n0 + wn + j * 16 + nl);
#pragma unroll
      for (int r = 0; r < 8; ++r)
        Cg[cbase + (size_t)r * MK] = acc[i][j][r];
    }
}

// ---------------------------------------------------------------------------
extern "C" void kernel_launch(void* const* d_in, const int* in_sizes, int n_in,
                              void* d_out, int out_size, void* d_ws, size_t ws_size,
                              hipStream_t stream) {
  (void)in_sizes; (void)n_in; (void)out_size; (void)ws_size;
  const float* a  = (const float*)d_in[0];
  const float* bm = (const float*)d_in[1];
  float* out   = (float*)d_out;
  float* flags = (float*)d_ws;   // 8*32*32 floats = 32 KB

  prescan_activity_kernel<<<dim3(NT, NT, NBATCH), 256, 0, stream>>>(a, flags);
  sparse_bmm_wmma_kernel<<<dim3(MK / BN, MK / BM, NBATCH), 256, 0, stream>>>(
      a, bm, flags, out);
}